// CombinedLoss_61864708932110
// MI455X (gfx1250) — compile-verified
//
#include <hip/hip_runtime.h>
#include <hip/hip_bf16.h>
#include <stdint.h>

#define INF_F 1000000.0f
#define BIGF  3.0e37f
#define NB3   512          // blocks for loss partial kernel
#define NELEM 4194304      // 16*512*512

typedef __attribute__((ext_vector_type(2))) float v2f;
typedef __attribute__((ext_vector_type(8))) float v8f;

#if defined(__HIP_DEVICE_COMPILE__) && defined(__gfx1250__) && __has_builtin(__builtin_amdgcn_wmma_f32_16x16x4_f32)
#define HAVE_WMMA 1
#else
#define HAVE_WMMA 0
#endif

// fast sigmoid: v_exp_f32 + v_rcp_f32 (avoid IEEE div expansion; loss-grade precision)
__device__ __forceinline__ float fast_sigmoid(float x) {
#if defined(__HIP_DEVICE_COMPILE__) && defined(__gfx1250__)
  return __builtin_amdgcn_rcpf(1.0f + __expf(-x));
#else
  return 1.0f / (1.0f + __expf(-x));
#endif
}

// ---------------- async global -> LDS helpers (CDNA5 async path) ------------
__device__ __forceinline__ void async_ld_b128(uint32_t lds_off, uint64_t gaddr) {
#if defined(__HIP_DEVICE_COMPILE__) && defined(__gfx1250__)
  asm volatile("global_load_async_to_lds_b128 %0, %1, off"
               :: "v"(lds_off), "v"(gaddr) : "memory");
#endif
}
__device__ __forceinline__ void wait_async_le4() {
#if defined(__HIP_DEVICE_COMPILE__) && defined(__gfx1250__)
  asm volatile("s_wait_asynccnt 4" ::: "memory");
#endif
}
__device__ __forceinline__ void wait_async_le0() {
#if defined(__HIP_DEVICE_COMPILE__) && defined(__gfx1250__)
  asm volatile("s_wait_asynccnt 0" ::: "memory");
#endif
}

// ============================================================================
// Kernel 1: exact 1D EDT along width, both site polarities, one wave32 per row
// fwd = x + cummin(pen - x), bwd = -x + rev_cummin(pen + x), g = min(fwd,bwd)
// ============================================================================
__device__ __forceinline__ void edt_row(const int (&t)[16], int lane, bool siteIsZero,
                                        float* __restrict__ outrow) {
  float p[16], fwd[16];
  // ---- forward inclusive cummin of (pen - x) ----
  float run = BIGF;
#pragma unroll
  for (int j = 0; j < 16; ++j) {
    float x   = (float)(lane * 16 + j);
    float pen = ((t[j] == 0) == siteIsZero) ? 0.0f : INF_F;
    run  = fminf(run, pen - x);
    p[j] = run;
  }
  float incl = run;
#pragma unroll
  for (int d = 1; d < 32; d <<= 1) {
    float o = __shfl_up(incl, d);
    if (lane >= d) incl = fminf(incl, o);
  }
  float excl = __shfl_up(incl, 1);
  if (lane == 0) excl = BIGF;
#pragma unroll
  for (int j = 0; j < 16; ++j) {
    float x = (float)(lane * 16 + j);
    fwd[j]  = x + fminf(p[j], excl);
  }
  // ---- backward inclusive cummin of (pen + x) ----
  run = BIGF;
#pragma unroll
  for (int j = 15; j >= 0; --j) {
    float x   = (float)(lane * 16 + j);
    float pen = ((t[j] == 0) == siteIsZero) ? 0.0f : INF_F;
    run  = fminf(run, pen + x);
    p[j] = run;
  }
  incl = run;
#pragma unroll
  for (int d = 1; d < 32; d <<= 1) {
    float o = __shfl_down(incl, d);
    if (lane < 32 - d) incl = fminf(incl, o);
  }
  excl = __shfl_down(incl, 1);
  if (lane == 31) excl = BIGF;
#pragma unroll
  for (int j = 0; j < 16; ++j) {
    float x = (float)(lane * 16 + j);
    float g = fminf(fwd[j], -x + fminf(p[j], excl));
    fwd[j]  = g * g;  // store g^2
  }
#pragma unroll
  for (int q = 0; q < 4; ++q) {
    float4 o = make_float4(fwd[4*q], fwd[4*q+1], fwd[4*q+2], fwd[4*q+3]);
    ((float4*)(outrow + lane * 16))[q] = o;
  }
}

__global__ __launch_bounds__(128) void row_edt_kernel(const int* __restrict__ targ,
                                                      float* __restrict__ g2in,
                                                      float* __restrict__ g2out,
                                                      int* __restrict__ fgcnt) {
  const int lane  = threadIdx.x & 31;
  const int gwave = blockIdx.x * 4 + (threadIdx.x >> 5);
  const int img   = gwave >> 9;
  const int row   = gwave & 511;
  const size_t rowoff = ((size_t)img * 512 + row) * 512;

  int t[16];
  const int4* rp = (const int4*)(targ + rowoff + lane * 16);
#pragma unroll
  for (int q = 0; q < 4; ++q) {
    int4 v = rp[q];
    t[4*q+0] = v.x; t[4*q+1] = v.y; t[4*q+2] = v.z; t[4*q+3] = v.w;
  }
  // per-image foreground count (deterministic int atomics)
  int c = 0;
#pragma unroll
  for (int j = 0; j < 16; ++j) c += (t[j] != 0);
  for (int off = 16; off; off >>= 1) c += __shfl_down(c, off);
  if (lane == 0 && c) atomicAdd(&fgcnt[img], c);

  // sites = ~mask (t==0) -> used for din ; sites = mask (t!=0) -> used for dout
  edt_row(t, lane, true,  g2in  + rowoff);
  edt_row(t, lane, false, g2out + rowoff);
}

// ============================================================================
// Kernel 2: column lower-envelope (brute force over y', as in reference),
// async double-buffered LDS staging, fused phi computation.
// Block = (img, 64-row ytile, 32-col xtile); loops y' in chunks of 64.
// ============================================================================
#define CH 64
__global__ __launch_bounds__(256) void envelope_kernel(const float* __restrict__ g2in,
                                                       const float* __restrict__ g2out,
                                                       const int* __restrict__ targ,
                                                       const int* __restrict__ fgcnt,
                                                       float* __restrict__ phi) {
  __shared__ float sm[2][2][CH * 32];   // [buf][in/out][yl*32+x]  = 32 KB
  const int xt  = blockIdx.x;           // 0..15
  const int yt  = blockIdx.y;           // 0..7
  const int img = blockIdx.z;           // 0..15
  const int tid  = threadIdx.x;
  const int xl   = tid & 31;
  const int yoff = tid >> 5;            // 0..7
  const int x0   = xt * 32;
  const size_t ibase = (size_t)img * 512 * 512;

  float accin[8], accout[8];
#pragma unroll
  for (int k = 0; k < 8; ++k) { accin[k] = BIGF; accout[k] = BIGF; }

  // loader mapping: b128 per lane, 8 lanes span one 32-float tile row
  const int lx4 = (tid & 7) * 4;        // 0,4,...,28
  const int lyl = tid >> 3;             // 0..31

  auto issue_chunk = [&](int c, int buf) {
#pragma unroll
    for (int r = 0; r < 2; ++r) {
      int yl = lyl + 32 * r;
      size_t go = (ibase + (size_t)(c * CH + yl) * 512 + x0 + lx4) * sizeof(float);
      async_ld_b128((uint32_t)(uintptr_t)&sm[buf][0][yl * 32 + lx4],
                    (uint64_t)(uintptr_t)g2in  + go);
      async_ld_b128((uint32_t)(uintptr_t)&sm[buf][1][yl * 32 + lx4],
                    (uint64_t)(uintptr_t)g2out + go);
    }
  };

  issue_chunk(0, 0);
  for (int c = 0; c < 8; ++c) {
    const int cur = c & 1;
    if (c < 7) { issue_chunk(c + 1, cur ^ 1); wait_async_le4(); }
    else       { wait_async_le0(); }
    __syncthreads();
    const float* bin  = sm[cur][0];
    const float* bout = sm[cur][1];
#pragma unroll 4
    for (int yl = 0; yl < CH; ++yl) {
      float gin  = bin [yl * 32 + xl];
      float gout = bout[yl * 32 + xl];
      float dy0  = (float)(yt * 64 + yoff - (c * CH + yl));
#pragma unroll
      for (int k = 0; k < 8; ++k) {
        float dy  = dy0 + (float)(8 * k);
        accin[k]  = fminf(accin[k],  fmaf(dy, dy, gin));
        accout[k] = fminf(accout[k], fmaf(dy, dy, gout));
      }
    }
    __syncthreads();
  }

  const bool anyfg = fgcnt[img] != 0;
#pragma unroll
  for (int k = 0; k < 8; ++k) {
    int y = yt * 64 + yoff + 8 * k;
    size_t idx = ibase + (size_t)y * 512 + x0 + xl;
    bool m   = targ[idx] != 0;
    float ph = m ? -sqrtf(accin[k]) : sqrtf(accout[k]);
    if (fabsf(ph) > 100000.0f) ph = 0.0f;
    if (!anyfg)                ph = 0.0f;
    phi[idx] = ph;
  }
}

// ============================================================================
// Kernel 3: fused sigmoid + 4 global partial sums. Memory-bound, so the adds
// ride the matrix pipe: v_wmma_f32_16x16x4_f32 with B = ones accumulates 64
// elements per wave per instruction into an f32 tile (row-sums), VALU stays
// free for address math / sigmoid. Trip count is uniform across the wave so
// EXEC is all-ones at every WMMA (ISA requirement).
// ============================================================================
__global__ __launch_bounds__(256) void loss_partial_kernel(const float* __restrict__ pred,
                                                           const int* __restrict__ targ,
                                                           const float* __restrict__ phi,
                                                           float* __restrict__ partials,
                                                           int n2) {
  const int tid  = threadIdx.x;
  const int lane = tid & 31;
  const int wv   = tid >> 5;
#if HAVE_WMMA
  v8f aP = {}, aT = {}, aI = {}, aB = {};
  const v2f ones = {1.0f, 1.0f};
#else
  float sP = 0.f, sT = 0.f, sI = 0.f, sB = 0.f;
#endif
  const int stride = gridDim.x * blockDim.x;
  for (int i = blockIdx.x * blockDim.x + tid; i < n2; i += stride) {
    float2 pv = ((const float2*)pred)[i];
    int2   tv = ((const int2*)  targ)[i];
    float2 fv = ((const float2*)phi )[i];
    float s0 = fast_sigmoid(pv.x);
    float s1 = fast_sigmoid(pv.y);
    float t0 = (float)tv.x, t1 = (float)tv.y;
#if HAVE_WMMA
    v2f vp = {s0, s1}, vt = {t0, t1}, vi = {s0 * t0, s1 * t1}, vb = {fv.x * s0, fv.y * s1};
    aP = __builtin_amdgcn_wmma_f32_16x16x4_f32(false, vp, false, ones, (short)0, aP, false, false);
    aT = __builtin_amdgcn_wmma_f32_16x16x4_f32(false, vt, false, ones, (short)0, aT, false, false);
    aI = __builtin_amdgcn_wmma_f32_16x16x4_f32(false, vi, false, ones, (short)0, aI, false, false);
    aB = __builtin_amdgcn_wmma_f32_16x16x4_f32(false, vb, false, ones, (short)0, aB, false, false);
#else
    sP += s0 + s1; sT += t0 + t1; sI += s0 * t0 + s1 * t1; sB += fv.x * s0 + fv.y * s1;
#endif
  }
#if HAVE_WMMA
  // D rows are row-sums replicated over 16 cols; grand total = col0 of all 16 rows
  // = (lane0: M=0..7) + (lane16: M=8..15) summed over 8 acc VGPRs.
  float sP = 0.f, sT = 0.f, sI = 0.f, sB = 0.f;
#pragma unroll
  for (int r = 0; r < 8; ++r) { sP += aP[r]; sT += aT[r]; sI += aI[r]; sB += aB[r]; }
  sP = __shfl(sP, 0) + __shfl(sP, 16);
  sT = __shfl(sT, 0) + __shfl(sT, 16);
  sI = __shfl(sI, 0) + __shfl(sI, 16);
  sB = __shfl(sB, 0) + __shfl(sB, 16);
#else
  for (int off = 16; off; off >>= 1) {
    sP += __shfl_down(sP, off); sT += __shfl_down(sT, off);
    sI += __shfl_down(sI, off); sB += __shfl_down(sB, off);
  }
#endif
  __shared__ float red[8][4];
  if (lane == 0) { red[wv][0] = sP; red[wv][1] = sT; red[wv][2] = sI; red[wv][3] = sB; }
  __syncthreads();
  if (tid == 0) {
    float a = 0.f, b = 0.f, c = 0.f, d = 0.f;
    for (int w = 0; w < 8; ++w) { a += red[w][0]; b += red[w][1]; c += red[w][2]; d += red[w][3]; }
    partials[blockIdx.x * 4 + 0] = a;
    partials[blockIdx.x * 4 + 1] = b;
    partials[blockIdx.x * 4 + 2] = c;
    partials[blockIdx.x * 4 + 3] = d;
  }
}

// ============================================================================
// Kernel 4: combine partials -> scalar loss
// ============================================================================
__global__ __launch_bounds__(256) void finalize_kernel(const float* __restrict__ partials,
                                                       float* __restrict__ out, int nblk) {
  __shared__ float red[256][4];
  float s[4] = {0.f, 0.f, 0.f, 0.f};
  for (int i = threadIdx.x; i < nblk; i += 256)
#pragma unroll
    for (int q = 0; q < 4; ++q) s[q] += partials[i * 4 + q];
#pragma unroll
  for (int q = 0; q < 4; ++q) red[threadIdx.x][q] = s[q];
  __syncthreads();
  for (int off = 128; off; off >>= 1) {
    if (threadIdx.x < (unsigned)off)
#pragma unroll
      for (int q = 0; q < 4; ++q) red[threadIdx.x][q] += red[threadIdx.x + off][q];
    __syncthreads();
  }
  if (threadIdx.x == 0) {
    float sumP = red[0][0], sumT = red[0][1], inter = red[0][2], sumBP = red[0][3];
    float dice     = 1.0f - (2.0f * inter + 1e-6f) / (sumP + sumT + 1e-6f);
    float boundary = sumBP / (float)NELEM;
    out[0] = 0.99f * dice + 0.01f * boundary;
  }
}

// ============================================================================
extern "C" void kernel_launch(void* const* d_in, const int* in_sizes, int n_in,
                              void* d_out, int out_size, void* d_ws, size_t ws_size,
                              hipStream_t stream) {
  const float* pred = (const float*)d_in[0];
  const int*   targ = (const int*)d_in[1];
  float* ws    = (float*)d_ws;
  float* g2in  = ws;                         // 16 MB
  float* g2out = ws + (size_t)NELEM;         // 16 MB
  float* phi   = ws + (size_t)2 * NELEM;     // 16 MB
  float* partials = ws + (size_t)3 * NELEM;  // NB3*4 floats
  int*   fgcnt = (int*)(ws + (size_t)3 * NELEM + 4 * NB3);

  hipMemsetAsync(fgcnt, 0, 16 * sizeof(int), stream);
  row_edt_kernel<<<2048, 128, 0, stream>>>(targ, g2in, g2out, fgcnt);
  envelope_kernel<<<dim3(16, 8, 16), 256, 0, stream>>>(g2in, g2out, targ, fgcnt, phi);
  loss_partial_kernel<<<NB3, 256, 0, stream>>>(pred, targ, phi, partials, NELEM / 2);
  finalize_kernel<<<1, 256, 0, stream>>>(partials, (float*)d_out, NB3);
}